// UpsampleUpFIRDn_6339371729661
// MI455X (gfx1250) — compile-verified
//
#include <hip/hip_runtime.h>
#include <stdint.h>

// ---------------------------------------------------------------------------
// upfirdn2d 2x upsample, separable 4-tap kernel k1 = [0.25, 0.75, 0.75, 0.25].
// x: (8,64,256,256) f32  ->  out: (8,64,512,512) f32
//   out[2n]   = 0.25*x[n-1] + 0.75*x[n]
//   out[2n+1] = 0.75*x[n]   + 0.25*x[n+1]        (zero pad; separable H and W)
// Memory bound: 128MB in + 512MB out @ 23.3 TB/s => ~27us floor (AI ~1.7 F/B),
// so no WMMA; use the CDNA5 TDM async tensor path for input staging, and
// non-temporal b128 stores so the 512MB output stream doesn't evict the
// (reused-as-halo) input tiles from the 192MB L2.
// ---------------------------------------------------------------------------

#define IN_H   256
#define IN_W   256
#define OUT_H  512
#define OUT_W  512
#define TILE_H 16      // input rows per block  -> 32 output rows
#define TILE_W 64      // input cols per block  -> 128 output cols
#define LDS_H  18      // TILE_H + 2 halo rows
#define LDS_W  72      // 64-col body at col 4 (TDM pads 8 dwords/row), halos at 3 / 68
#define BODY   4

typedef unsigned int u32x4 __attribute__((ext_vector_type(4)));
typedef int          i32x8 __attribute__((ext_vector_type(8)));
typedef int          i32x4 __attribute__((ext_vector_type(4)));
typedef float        f32x4 __attribute__((ext_vector_type(4)));   // NT-store friendly

__global__ __launch_bounds__(256) void upfirdn_up2(const float* __restrict__ x,
                                                   float* __restrict__ out)
{
    __shared__ float s[LDS_H][LDS_W];

    const int tid   = threadIdx.x;
    const int C     = blockIdx.x * TILE_W;   // input col base
    const int R     = blockIdx.y * TILE_H;   // input row base
    const int plane = blockIdx.z;            // N*C planes

    const float* xp = x   + (size_t)plane * IN_H * IN_W;
    float*       op = out + (size_t)plane * OUT_H * OUT_W;

    const int rowStart = (R == 0) ? 0 : (R - 1);
    const int rowEnd   = (R + TILE_H < IN_H) ? (R + TILE_H) : (IN_H - 1);
    const int nrows    = rowEnd - rowStart + 1;   // 17 (edge tiles) or 18
    const int ldsRow0  = rowStart - (R - 1);      // 1 for top tile, else 0

#if defined(__gfx1250__) && __has_builtin(__builtin_amdgcn_tensor_load_to_lds)
    // ---- TDM: DMA the in-bounds 64 x nrows body straight into LDS ----------
    // D# group0: count=1 | lds_addr | global_addr(57b) | type=2
    // D# group1: data_size=4B, pad_enable, pad_interval=64dw (code 5),
    //            pad_amount=8dw (code 7)  => LDS row stride 72 dwords.
    //            tensor_dim0=64, tensor_dim1=nrows (exact => never OOB),
    //            tile_dim0=64, tile_dim1=nrows, dim0 stride = 256 elements.
    if (tid < 32) {   // one wave issues the descriptor (wave-level DMA)
        const uint32_t lds_off = (uint32_t)(uintptr_t)&s[ldsRow0][BODY];
        const uint64_t gaddr   = (uint64_t)(uintptr_t)(xp + (size_t)rowStart * IN_W + C);
        const u32x4 g0 = { 1u,
                           lds_off,
                           (uint32_t)gaddr,
                           (uint32_t)((gaddr >> 32) & 0x1FFFFFFu) | (2u << 30) };
        const i32x8 g1 = { (int)((2u << 16) | (1u << 20) | (5u << 22) | (7u << 25)),
                           (int)(64u << 16),                 // tensor_dim0 lo16 -> [31:16]
                           (int)((uint32_t)nrows << 16),     // tensor_dim1 lo16 -> [31:16]
                           (int)(64u << 16),                 // tile_dim0 -> [31:16]
                           nrows,                            // tile_dim1 (tile_dim2 = 0)
                           IN_W,                             // tensor_dim0_stride lo32
                           0, 0 };
        const i32x4 g2 = { 0, 0, 0, 0 };
        const i32x4 g3 = { 0, 0, 0, 0 };
        const i32x8 g4 = { 0, 0, 0, 0, 0, 0, 0, 0 };   // unused trailing group (6-arg form)
        __builtin_amdgcn_tensor_load_to_lds(g0, g1, g2, g3, g4, 0);
    }
    // Rows the TDM skipped (outside the image) must read as zero padding.
    if (R == 0             && tid >= 64  && tid < 128) s[0]      [BODY + (tid -  64)] = 0.0f;
    if (R == IN_H - TILE_H && tid >= 128 && tid < 192) s[LDS_H-1][BODY + (tid - 128)] = 0.0f;
#else
    // Fallback: cooperative load of the 18x64 body with zero padding.
    for (int e = tid; e < LDS_H * TILE_W; e += 256) {
        const int i = e >> 6, j = e & 63;
        const int gr = R - 1 + i;
        s[i][BODY + j] = (gr >= 0 && gr < IN_H) ? xp[(size_t)gr * IN_W + C + j] : 0.0f;
    }
#endif
    // Halo columns (never overlap the TDM write region).
    if (tid < LDS_H) {
        const int gr = R - 1 + tid;
        s[tid][BODY - 1] =
            (gr >= 0 && gr < IN_H && C > 0) ? xp[(size_t)gr * IN_W + (C - 1)] : 0.0f;
    } else if (tid >= 32 && tid < 32 + LDS_H) {
        const int i = tid - 32, gr = R - 1 + i;
        s[i][BODY + TILE_W] =
            (gr >= 0 && gr < IN_H && C + TILE_W < IN_W) ? xp[(size_t)gr * IN_W + C + TILE_W]
                                                        : 0.0f;
    }
#if defined(__gfx1250__) && __has_builtin(__builtin_amdgcn_s_wait_tensorcnt)
    __builtin_amdgcn_s_wait_tensorcnt(0);   // no-op for waves that issued nothing
#endif
    __syncthreads();

    // ---- Compute: each iteration = 2 output rows x 4 output cols -----------
    const float WL = 0.25f, WH = 0.75f;
#pragma unroll
    for (int it = 0; it < 2; ++it) {
        const int idx = tid + 256 * it;      // 0..511 micro-tiles
        const int r   = idx >> 5;            // 0..15 : input row in tile
        const int p   = idx & 31;            // 0..31 : input col pair
        const int kc  = 2 * p;
        const float* s0 = &s[r    ][BODY + kc];   // row R+r-1
        const float* s1 = &s[r + 1][BODY + kc];   // row R+r
        const float* s2 = &s[r + 2][BODY + kc];   // row R+r+1

        f32x4 h0, h1, h2;
        { const float a=s0[-1], b=s0[0], c=s0[1], d=s0[2];
          h0.x=WL*a+WH*b; h0.y=WH*b+WL*c; h0.z=WL*b+WH*c; h0.w=WH*c+WL*d; }
        { const float a=s1[-1], b=s1[0], c=s1[1], d=s1[2];
          h1.x=WL*a+WH*b; h1.y=WH*b+WL*c; h1.z=WL*b+WH*c; h1.w=WH*c+WL*d; }
        { const float a=s2[-1], b=s2[0], c=s2[1], d=s2[2];
          h2.x=WL*a+WH*b; h2.y=WH*b+WL*c; h2.z=WL*b+WH*c; h2.w=WH*c+WL*d; }

        const f32x4 oe = WL * h0 + WH * h1;   // even output row (elementwise)
        const f32x4 oo = WH * h1 + WL * h2;   // odd  output row (elementwise)

        const int orow = 2 * (R + r);
        const int ocol = 2 * C + 4 * p;                       // 16B aligned
        f32x4* d0 = (f32x4*)(op + (size_t)orow       * OUT_W + ocol);
        f32x4* d1 = (f32x4*)(op + (size_t)(orow + 1) * OUT_W + ocol);
        __builtin_nontemporal_store(oe, d0);   // global_store_b128 th:NT (streaming)
        __builtin_nontemporal_store(oo, d1);   // global_store_b128 th:NT (streaming)
    }
}

extern "C" void kernel_launch(void* const* d_in, const int* in_sizes, int n_in,
                              void* d_out, int out_size, void* d_ws, size_t ws_size,
                              hipStream_t stream)
{
    const float* x   = (const float*)d_in[0];   // (8,64,256,256) f32
    // d_in[1] = 4x4 kernel (separable [.25,.75,.75,.25]^T x same, hardcoded),
    // d_in[2] = factor (=2, hardcoded in the polyphase derivation).
    float* out = (float*)d_out;                 // (8,64,512,512) f32

    const int planes = in_sizes[0] / (IN_H * IN_W);   // 8*64 = 512
    dim3 grid(IN_W / TILE_W, IN_H / TILE_H, planes);  // (4, 16, 512)
    upfirdn_up2<<<grid, 256, 0, stream>>>(x, out);
}